// SelectTopk_45655502356783
// MI455X (gfx1250) — compile-verified
//
#include <hip/hip_runtime.h>
#include <hip/hip_bf16.h>
#include <math.h>

#ifndef __has_builtin
#define __has_builtin(x) 0
#endif

#define E_LOGICAL 256          // logical experts per token (row length)
#define TOPK 8
#define MAXREP 2               // replicas per logical expert
#define WAVE 32                // gfx1250 is wave32-only
#define WAVES_PER_BLOCK 8
#define TOKENS_PER_WAVE 4
#define BLOCK_THREADS (WAVES_PER_BLOCK * WAVE)            // 256
#define TOKENS_PER_BLOCK (WAVES_PER_BLOCK * TOKENS_PER_WAVE)  // 32

#if __has_builtin(__builtin_amdgcn_global_load_async_to_lds_b32)
#define HAVE_ASYNC_LDS 1
#else
#define HAVE_ASYNC_LDS 0
#endif

#if HAVE_ASYNC_LDS
#if __has_builtin(__builtin_amdgcn_s_wait_asynccnt)
#define WAIT_ASYNC() __builtin_amdgcn_s_wait_asynccnt(0)
#else
#define WAIT_ASYNC() asm volatile("s_wait_asynccnt 0" ::: "memory")
#endif
typedef __attribute__((address_space(1))) int* gptr_i32;
typedef __attribute__((address_space(3))) int* lptr_i32;
#endif

typedef float f4 __attribute__((ext_vector_type(4)));

// ---- wave32 butterfly reductions via DS_SWIZZLE_B32 (group-of-32 xor mode) ----
template <int M>
__device__ __forceinline__ int swz32(int v) {
  // offset[15]=0 -> group-of-32: {xor[14:10], or[9:5], and[4:0]}
  return __builtin_amdgcn_ds_swizzle(v, 0x1f | (M << 10));
}
template <int M>
__device__ __forceinline__ float swzf(float v) {
  return __int_as_float(swz32<M>(__float_as_int(v)));
}

__device__ __forceinline__ float waveSumF(float v) {
  v += swzf<1>(v);
  v += swzf<2>(v);
  v += swzf<4>(v);
  v += swzf<8>(v);
  v += swzf<16>(v);
  return v;
}
__device__ __forceinline__ unsigned waveMaxU32(unsigned v) {
  unsigned o;
  o = (unsigned)swz32<1>((int)v);  v = (o > v) ? o : v;
  o = (unsigned)swz32<2>((int)v);  v = (o > v) ? o : v;
  o = (unsigned)swz32<4>((int)v);  v = (o > v) ? o : v;
  o = (unsigned)swz32<8>((int)v);  v = (o > v) ? o : v;
  o = (unsigned)swz32<16>((int)v); v = (o > v) ? o : v;
  return v;
}

// Monotone float->u32 key: preserves IEEE-754 total order for normal values.
__device__ __forceinline__ unsigned orderKey(float f) {
  unsigned b = __float_as_uint(f);
  return (b & 0x80000000u) ? ~b : (b | 0x80000000u);
}
__device__ __forceinline__ float keyToFloat(unsigned k) {
  unsigned b = (k & 0x80000000u) ? (k ^ 0x80000000u) : ~k;
  return __uint_as_float(b);
}

__global__ __launch_bounds__(BLOCK_THREADS)
void MoeTopkRemap_kernel(const float* __restrict__ logits,    // [T, 256]
                         const int* __restrict__ log2phy,     // [256, MAXREP]
                         const int* __restrict__ logic_cnt,   // [256]
                         const int* __restrict__ ep_rank_ptr, // [1]
                         int num_tokens,
                         int* __restrict__ out_ids,           // [T, 8]
                         float* __restrict__ out_w) {         // [T, 8]
  __shared__ int s_l2p[E_LOGICAL * MAXREP];
  __shared__ int s_cnt[E_LOGICAL];

  const int tid = threadIdx.x;

  // ---- Stage remap tables into LDS once per block (gfx1250 async path) ----
#if HAVE_ASYNC_LDS
  for (int i = tid; i < E_LOGICAL * MAXREP; i += BLOCK_THREADS)
    __builtin_amdgcn_global_load_async_to_lds_b32(
        (gptr_i32)(log2phy + i), (lptr_i32)(s_l2p + i), 0, 0);
  for (int i = tid; i < E_LOGICAL; i += BLOCK_THREADS)
    __builtin_amdgcn_global_load_async_to_lds_b32(
        (gptr_i32)(logic_cnt + i), (lptr_i32)(s_cnt + i), 0, 0);
  WAIT_ASYNC();
#else
  for (int i = tid; i < E_LOGICAL * MAXREP; i += BLOCK_THREADS) s_l2p[i] = log2phy[i];
  for (int i = tid; i < E_LOGICAL; i += BLOCK_THREADS) s_cnt[i] = logic_cnt[i];
#endif
  __syncthreads();

  const int lane = tid & (WAVE - 1);
  const int wave = tid >> 5;
  const int ep = ep_rank_ptr[0];
  const int tok0 = (int)blockIdx.x * TOKENS_PER_BLOCK + wave * TOKENS_PER_WAVE;

  for (int tt = 0; tt < TOKENS_PER_WAVE; ++tt) {
    const int t = tok0 + tt;
    if (t >= num_tokens) break;  // uniform per wave
    const float* row = logits + (size_t)t * E_LOGICAL;

    // Prefetch next token's row while we crunch this one (global_prefetch_b8).
    if (tt + 1 < TOKENS_PER_WAVE && t + 1 < num_tokens)
      __builtin_prefetch(row + E_LOGICAL + lane * 8, 0, 0);

    // 256 logits / 32 lanes = 8 per lane via two b128 nontemporal loads
    // (streamed once -> keep it out of L2 so the tables stay hot).
    f4 a = __builtin_nontemporal_load((const f4*)(row + 4 * lane));
    f4 b = __builtin_nontemporal_load((const f4*)(row + (E_LOGICAL / 2) + 4 * lane));

    // Pack: value key in [63:32], (255 - global_idx) in [7:0].
    // u64 compare => order by value, ties -> smaller index (jax.lax.top_k).
    // Element j of 'a' has idx 4*lane+j; of 'b' has idx 128+4*lane+j.
    unsigned long long cand[8];
#pragma unroll
    for (int j = 0; j < 4; ++j)
      cand[j] = ((unsigned long long)orderKey(a[j]) << 32) |
                (unsigned)(255 - (4 * lane + j));
#pragma unroll
    for (int j = 0; j < 4; ++j)
      cand[4 + j] = ((unsigned long long)orderKey(b[j]) << 32) |
                    (unsigned)(127 - (4 * lane + j));

    // Lane-local descending sort: Batcher odd-even network, 19 comparators.
#define CEX(i, j)                                            \
    {                                                        \
      unsigned long long x_ = cand[i], y_ = cand[j];         \
      const bool g_ = (x_ > y_);                             \
      cand[i] = g_ ? x_ : y_;                                \
      cand[j] = g_ ? y_ : x_;                                \
    }
    CEX(0, 1) CEX(2, 3) CEX(4, 5) CEX(6, 7)
    CEX(0, 2) CEX(1, 3) CEX(4, 6) CEX(5, 7)
    CEX(1, 2) CEX(5, 6)
    CEX(0, 4) CEX(1, 5) CEX(2, 6) CEX(3, 7)
    CEX(2, 4) CEX(3, 5)
    CEX(1, 2) CEX(3, 4) CEX(5, 6)
#undef CEX

    // ---- Round 0: wave argmax of heads; winner value == row max. ----
    unsigned key0 = (unsigned)(cand[0] >> 32);
    unsigned wkey = waveMaxU32(key0);
    unsigned enc0 = (key0 == wkey) ? (unsigned)cand[0] : 0u;
    unsigned wenc = waveMaxU32(enc0);   // max(255-idx) = min index
    const float rowmax = keyToFloat(wkey);
    {
      const unsigned long long w =
          ((unsigned long long)wkey << 32) | (unsigned long long)wenc;
      if (cand[0] == w) {  // unique owner pops its head
        cand[0] = cand[1]; cand[1] = cand[2]; cand[2] = cand[3];
        cand[3] = cand[4]; cand[4] = cand[5]; cand[5] = cand[6];
        cand[6] = cand[7]; cand[7] = 0ull;
      }
    }
    float myVal = rowmax;            // correct for lane 0 (slot 0)
    int myIdx = 255 - (int)wenc;     // lanes 1..7 overwritten below

    // Softmax denominator (v_exp_f32 is TRANS32 -> co-issues with VALU).
    float lsum = 0.f;
#pragma unroll
    for (int j = 0; j < 4; ++j) lsum += __expf(a[j] - rowmax);
#pragma unroll
    for (int j = 0; j < 4; ++j) lsum += __expf(b[j] - rowmax);
    const float invsum = 1.0f / waveSumF(lsum);

    // ---- Rounds 1..7 ----
#pragma unroll
    for (int s = 1; s < TOPK; ++s) {
      key0 = (unsigned)(cand[0] >> 32);
      wkey = waveMaxU32(key0);
      enc0 = (key0 == wkey) ? (unsigned)cand[0] : 0u;
      wenc = waveMaxU32(enc0);
      const unsigned long long w =
          ((unsigned long long)wkey << 32) | (unsigned long long)wenc;
      if (cand[0] == w) {
        cand[0] = cand[1]; cand[1] = cand[2]; cand[2] = cand[3];
        cand[3] = cand[4]; cand[4] = cand[5]; cand[5] = cand[6];
        cand[6] = cand[7]; cand[7] = 0ull;
      }
      if (lane == s) {  // lane s owns output slot s (descending order)
        myIdx = 255 - (int)wenc;
        myVal = keyToFloat(wkey);
      }
    }

    if (lane < TOPK) {
      const float wgt = __expf(myVal - rowmax) * invsum;
      int c = s_cnt[myIdx];
      if (c < 1) c = 1;
      const int rep = (c > 1) ? ((t + ep) % c) : 0;
      const int phy = s_l2p[myIdx * MAXREP + rep];
      const size_t o = (size_t)t * TOPK + lane;
      __builtin_nontemporal_store(phy, out_ids + o);
      __builtin_nontemporal_store(wgt, out_w + o);
    }
  }
}

extern "C" void kernel_launch(void* const* d_in, const int* in_sizes, int n_in,
                              void* d_out, int out_size, void* d_ws, size_t ws_size,
                              hipStream_t stream) {
  // setup_inputs() order:
  // 0: router_logits_fp32 [T,256] f32
  // 1: topk_ids buffer    [T,8]  i32 (shape carrier, unused)
  // 2: topk_weights buf   [T,8]  f32 (unused)
  // 3: log2phy            [256,MAXREP] i32
  // 4: logic_expert_cnt   [256]  i32
  // 5: phy_exp_num scalar (unused in math)
  // 6: ep_rank scalar     i32 (device pointer, read in-kernel)
  const float* logits = (const float*)d_in[0];
  const int* log2phy = (const int*)d_in[3];
  const int* logic_cnt = (const int*)d_in[4];
  const int* ep_rank = (const int*)d_in[6];

  const int E = in_sizes[4];                 // 256 logical experts
  const int T = in_sizes[0] / E;             // num tokens
  const int K = in_sizes[1] / T;             // top-k (== 8)

  // d_out = concat(phy_ids [T,K] i32, weights [T,K] f32), flat.
  int* out_ids = (int*)d_out;
  float* out_w = (float*)d_out + (size_t)T * K;

  dim3 block(BLOCK_THREADS);
  dim3 grid((T + TOKENS_PER_BLOCK - 1) / TOKENS_PER_BLOCK);
  MoeTopkRemap_kernel<<<grid, block, 0, stream>>>(logits, log2phy, logic_cnt,
                                                  ep_rank, T, out_ids, out_w);
}